// ContrastiveLoss_3925600108857
// MI455X (gfx1250) — compile-verified
//
#include <hip/hip_runtime.h>

// Contrastive pair loss, MI455X (gfx1250, wave32).
// Memory-bound: 512 MiB stream of X -> ~23 us floor at 23.3 TB/s.
// Row reduction done with V_WMMA_F32_16X16X4_F32 against an all-ones B
// (D[m][n] = sum_k A[m][k]): 16 row-sums per matrix op, f32 throughout.

typedef float v2f __attribute__((ext_vector_type(2)));
typedef float v4f __attribute__((ext_vector_type(4)));
typedef float v8f __attribute__((ext_vector_type(8)));

#define HDIM   128   // hidden size H
#define ROWLEN 256   // 2*H floats per pair row
#define TILE_M 16    // pairs per wave
#define WAVES_PER_BLOCK 8

__global__ __launch_bounds__(256)
void contrastive_loss_wmma(const float* __restrict__ X,
                           const int*   __restrict__ y,
                           float* __restrict__ out,
                           int npairs)
{
    const int lane = threadIdx.x & 31;
    const int wave = threadIdx.x >> 5;
    const int tile = blockIdx.x * WAVES_PER_BLOCK + wave;
    const int pair0 = tile * TILE_M;
    if (pair0 >= npairs) return;           // wave-uniform guard (EXEC stays all-1s)

    const int m = lane & 15;               // row M within tile (A-matrix layout)
    const int h = lane >> 4;               // half-wave: supplies K-slots {2h, 2h+1}

    // Lane streams 4 consecutive columns per step; half 0 covers cols 0..63,
    // half 1 covers cols 64..127 of its row. Sum over K is order-independent.
    const float* rowA = X + (size_t)(pair0 + m) * ROWLEN + 64 * h;  // e_a
    const float* rowB = rowA + HDIM;                                // e_b

    v8f acc0 = {0.f, 0.f, 0.f, 0.f, 0.f, 0.f, 0.f, 0.f};
    v8f acc1 = acc0;
    const v2f ones = {1.0f, 1.0f};         // B matrix: 4x16 of 1.0 -> row-sum

    #pragma unroll
    for (int t = 0; t < 16; ++t) {
        v4f a = *(const v4f*)(rowA + 4 * t);
        v4f b = *(const v4f*)(rowB + 4 * t);
        v4f d = a - b;
        v2f s0 = { d.x * d.x, d.y * d.y }; // K-slots {2h, 2h+1} of WMMA #0
        v2f s1 = { d.z * d.z, d.w * d.w }; // K-slots {2h, 2h+1} of WMMA #1
        // 8-arg form: (neg_a, A, neg_b, B, c_mod, C, reuse_a, reuse_b)
        acc0 = __builtin_amdgcn_wmma_f32_16x16x4_f32(false, s0, false, ones,
                                                     (short)0, acc0, false, false);
        acc1 = __builtin_amdgcn_wmma_f32_16x16x4_f32(false, s1, false, ones,
                                                     (short)0, acc1, false, false);
    }

    // C/D layout: VGPR r, lanes 0-15 hold M=r (N=lane), lanes 16-31 hold M=8+r.
    // Every column is identical (B was all-ones), so read column N=0:
    // lane 0 owns dist2 for pairs pair0+0..7, lane 16 owns pair0+8..15.
    if (m == 0) {
        const int base = pair0 + 8 * h;
        #pragma unroll
        for (int r = 0; r < 8; ++r) {
            float d2  = acc0[r] + acc1[r];
            int   adj = y[base + r];
            float loss = (adj == 1) ? d2
                       : ((adj == 0) ? fmaxf(1.0f - d2, 0.0f) : 0.0f);
            out[base + r] = loss;
        }
    }
}

extern "C" void kernel_launch(void* const* d_in, const int* in_sizes, int n_in,
                              void* d_out, int out_size, void* d_ws, size_t ws_size,
                              hipStream_t stream) {
    const float* X   = (const float*)d_in[0];   // (B, N*N, 2H) f32
    const int*   y   = (const int*)  d_in[1];   // (B, N, N)   i32
    float*       out = (float*)d_out;           // (B, N*N)    f32

    const int npairs = in_sizes[1];             // B*N*N = 524288 (multiple of 16)
    const int tiles  = (npairs + TILE_M - 1) / TILE_M;
    const int blocks = (tiles + WAVES_PER_BLOCK - 1) / WAVES_PER_BLOCK;

    hipLaunchKernelGGL(contrastive_loss_wmma, dim3(blocks), dim3(256), 0, stream,
                       X, y, out, npairs);
}